// ResidualBlock_32152125178170
// MI455X (gfx1250) — compile-verified
//
#include <hip/hip_runtime.h>
#include <math.h>

// Problem constants from the reference
#define NN 100000
#define EE 600000
#define DD 128
#define GG 256
#define EPSV 1e-5f

typedef float v2f __attribute__((ext_vector_type(2)));
typedef float v8f __attribute__((ext_vector_type(8)));

// ---------------------------------------------------------------------------
// 0) init: agg = -inf, stats = 0  (re-done every launch: graph replay safe)
// ---------------------------------------------------------------------------
__global__ __launch_bounds__(256) void init_kernel(float* __restrict__ agg,
                                                   float* __restrict__ sumA,
                                                   float* __restrict__ sqA,
                                                   float* __restrict__ sumB,
                                                   float* __restrict__ sqB,
                                                   float* __restrict__ cnt) {
  int idx = blockIdx.x * 256 + threadIdx.x;
  if (idx < NN * DD) agg[idx] = -INFINITY;
  if (idx < GG * DD) {
    sumA[idx] = 0.f; sqA[idx] = 0.f; sumB[idx] = 0.f; sqB[idx] = 0.f;
  }
  if (idx < GG) cnt[idx] = 0.f;
}

// ---------------------------------------------------------------------------
// 1) per-graph sum / sumsq / count for x.  batch is sorted -> run-length
//    compact locally, flush one atomic per (graph-run, col).
//    Block covers 64 rows x 128 cols with 256 threads (each: 32 rows, 1 col).
// ---------------------------------------------------------------------------
__global__ __launch_bounds__(256) void stats_kernel(const float* __restrict__ x,
                                                    const int* __restrict__ batch,
                                                    float* __restrict__ sum,
                                                    float* __restrict__ sq,
                                                    float* __restrict__ cnt) {
  const int col = threadIdx.x & 127;
  const int half = threadIdx.x >> 7;
  const int row0 = blockIdx.x * 64;
  float s = 0.f, q = 0.f, rc = 0.f;
  int curg = -1;
  for (int i = 0; i < 32; ++i) {
    int r = row0 + 2 * i + half;
    if (r >= NN) break;
    int g = batch[r];
    if (g != curg) {
      if (curg >= 0) {
        atomicAdd(&sum[curg * DD + col], s);
        atomicAdd(&sq[curg * DD + col], q);
        if (col == 0) atomicAdd(&cnt[curg], rc);
      }
      curg = g; s = 0.f; q = 0.f; rc = 0.f;
    }
    float v = x[(size_t)r * DD + col];
    s += v; q += v * v; rc += 1.f;
  }
  if (curg >= 0) {
    atomicAdd(&sum[curg * DD + col], s);
    atomicAdd(&sq[curg * DD + col], q);
    if (col == 0) atomicAdd(&cnt[curg], rc);
  }
}

// ---------------------------------------------------------------------------
// 2) finalize per-graph stats:  shift = mean*mean_scale,
//    var = E[x^2] - mean^2*(2*ms - ms^2)   (one-pass identity)
// ---------------------------------------------------------------------------
__global__ __launch_bounds__(256) void finalize_kernel(const float* __restrict__ sum,
                                                       const float* __restrict__ sq,
                                                       const float* __restrict__ cnt,
                                                       const float* __restrict__ ms,
                                                       float* __restrict__ meanOut,
                                                       float* __restrict__ invOut) {
  int idx = blockIdx.x * 256 + threadIdx.x;   // < G*D exactly
  int g = idx >> 7, col = idx & 127;
  float c = cnt[g]; c = c > 1.f ? c : 1.f;
  float m = sum[idx] / c;
  float ex2 = sq[idx] / c;
  float a = ms[col];
  float var = ex2 - m * m * (2.f * a - a * a);
  var = var > 0.f ? var : 0.f;
  meanOut[idx] = m * a;
  invOut[idx] = rsqrtf(var + EPSV);
}

// ---------------------------------------------------------------------------
// 3) apply graph norm:  out = w*(in - shift[g])*inv[g] + b
// ---------------------------------------------------------------------------
__global__ __launch_bounds__(256) void norm_apply_kernel(const float* __restrict__ in,
                                                         const int* __restrict__ batch,
                                                         const float* __restrict__ mean,
                                                         const float* __restrict__ inv,
                                                         const float* __restrict__ w,
                                                         const float* __restrict__ b,
                                                         float* __restrict__ out) {
  int idx = blockIdx.x * 256 + threadIdx.x;   // < N*D exactly
  int row = idx >> 7, col = idx & 127;
  int g = batch[row];
  float v = (in[idx] - mean[g * DD + col]) * inv[g * DD + col];
  out[idx] = w[col] * v + b[col];
}

// ---------------------------------------------------------------------------
// 4) scatter-max over edges. One wave per edge, float4 per lane.
//    Sign-split int/uint atomic trick -> single global_atomic_max_i32 /
//    global_atomic_min_u32 per element (works with -inf init, finite data).
// ---------------------------------------------------------------------------
__device__ __forceinline__ void atomicMaxFloat(float* addr, float v) {
  if (v >= 0.f) atomicMax((int*)addr, __float_as_int(v));
  else          atomicMin((unsigned int*)addr, __float_as_uint(v));
}

__global__ __launch_bounds__(256) void scatter_max_kernel(const float* __restrict__ h1,
                                                          const int* __restrict__ src,
                                                          const int* __restrict__ dst,
                                                          float* __restrict__ agg) {
  int tid = blockIdx.x * 256 + threadIdx.x;
  int e = tid >> 5;               // wave-uniform: scalar load of src/dst
  int c4 = (tid & 31) * 4;        // lane covers 4 contiguous features
  int s = src[e], d = dst[e];
  const float4 v = *(const float4*)(h1 + (size_t)s * DD + c4);
  float* o = agg + (size_t)d * DD + c4;
  atomicMaxFloat(o + 0, v.x);
  atomicMaxFloat(o + 1, v.y);
  atomicMaxFloat(o + 2, v.z);
  atomicMaxFloat(o + 3, v.w);
}

// ---------------------------------------------------------------------------
// 5) fused:  h2 = relu(x + agg@w_l^T + b_l + h1@w_r^T)  + norm-2 stats.
//    One wave computes one 16x16 output tile via V_WMMA_F32_16X16X4_F32,
//    block = 8 waves = 16 rows x full 128 cols.  EXEC is all-ones (exact grid).
//    A 16x4 f32 layout : lane m=l&15, K-pair = (l>>4)*2 + {0,1}  (2 VGPRs)
//    B 4x16 f32 layout : lane n=l&15, K-pair = (l>>4)*2 + {0,1}  (mirror)
//    C/D 16x16 f32     : VGPR j -> row j (lanes 0-15) / row j+8 (lanes 16-31)
// ---------------------------------------------------------------------------
__global__ __launch_bounds__(256) void gemm_res_relu_stats(const float* __restrict__ agg,
                                                           const float* __restrict__ h1,
                                                           const float* __restrict__ x,
                                                           const int* __restrict__ batch,
                                                           const float* __restrict__ w_l,
                                                           const float* __restrict__ b_l,
                                                           const float* __restrict__ w_r,
                                                           float* __restrict__ h2,
                                                           float* __restrict__ sumB,
                                                           float* __restrict__ sqB) {
  const int lane = threadIdx.x & 31;
  const int wave = threadIdx.x >> 5;
  const int row_base = blockIdx.x * 16;   // N = 100000 = 6250 * 16, exact
  const int col_base = wave * 16;         // 8 waves * 16 = 128 cols
  const int m = lane & 15;
  const int koff = (lane >> 4) * 2;

  const float* aA = agg + (size_t)(row_base + m) * DD + koff;
  const float* aH = h1  + (size_t)(row_base + m) * DD + koff;
  const float* bL = w_l + (size_t)(col_base + m) * DD + koff;  // B[k][n] = w_l[n][k]
  const float* bR = w_r + (size_t)(col_base + m) * DD + koff;

  v8f c = {};
#pragma unroll
  for (int kk = 0; kk < DD; kk += 4) {
    v2f a0 = *(const v2f*)(aA + kk);
    // PyG semantics: empty segments (-inf after max) behave as 0
    a0.x = (a0.x == -INFINITY) ? 0.f : a0.x;
    a0.y = (a0.y == -INFINITY) ? 0.f : a0.y;
    v2f b0 = *(const v2f*)(bL + kk);
    c = __builtin_amdgcn_wmma_f32_16x16x4_f32(false, a0, false, b0, (short)0, c,
                                              false, false);
    v2f a1 = *(const v2f*)(aH + kk);
    v2f b1 = *(const v2f*)(bR + kk);
    c = __builtin_amdgcn_wmma_f32_16x16x4_f32(false, a1, false, b1, (short)0, c,
                                              false, false);
  }

  const int col = col_base + (lane & 15);
  const float bias = b_l[col];
#pragma unroll
  for (int j = 0; j < 8; ++j) {
    int row = row_base + j + 8 * (lane >> 4);
    float v = c[j] + bias + x[(size_t)row * DD + col];
    v = v > 0.f ? v : 0.f;                 // relu(x + h)
    h2[(size_t)row * DD + col] = v;
    int g = batch[row];
    atomicAdd(&sumB[g * DD + col], v);
    atomicAdd(&sqB[g * DD + col], v * v);
  }
}

// ---------------------------------------------------------------------------
extern "C" void kernel_launch(void* const* d_in, const int* in_sizes, int n_in,
                              void* d_out, int out_size, void* d_ws, size_t ws_size,
                              hipStream_t stream) {
  const float* x     = (const float*)d_in[0];
  const int*   ei    = (const int*)d_in[1];     // [2,E] flat
  const int*   batch = (const int*)d_in[2];
  const float* n1_w  = (const float*)d_in[3];
  const float* n1_b  = (const float*)d_in[4];
  const float* n1_ms = (const float*)d_in[5];
  const float* n2_w  = (const float*)d_in[6];
  const float* n2_b  = (const float*)d_in[7];
  const float* n2_ms = (const float*)d_in[8];
  const float* w_l   = (const float*)d_in[9];
  const float* b_l   = (const float*)d_in[10];
  const float* w_r   = (const float*)d_in[11];

  const int* srcI = ei;
  const int* dstI = ei + EE;

  const size_t ND = (size_t)NN * DD;
  const size_t GD = (size_t)GG * DD;
  float* ws    = (float*)d_ws;
  float* h1    = ws;                 // N*D
  float* agg   = h1 + ND;            // N*D
  float* h2    = agg + ND;           // N*D
  float* sumA  = h2 + ND;            // G*D
  float* sqA   = sumA + GD;          // G*D
  float* sumB  = sqA + GD;           // G*D
  float* sqB   = sumB + GD;          // G*D
  float* meanA = sqB + GD;           // G*D
  float* invA  = meanA + GD;         // G*D
  float* meanB = invA + GD;          // G*D
  float* invB  = meanB + GD;         // G*D
  float* cnt   = invB + GD;          // G

  const int blkND   = (int)(ND / 256);        // 50000, exact
  const int blkStat = (NN + 63) / 64;         // 1563
  const int blkGD   = (int)(GD / 256);        // 128, exact
  const int blkEdge = (EE * 32) / 256;        // 75000, exact
  const int blkGemm = NN / 16;                // 6250, exact

  init_kernel<<<blkND, 256, 0, stream>>>(agg, sumA, sqA, sumB, sqB, cnt);

  // GraphNorm 1
  stats_kernel<<<blkStat, 256, 0, stream>>>(x, batch, sumA, sqA, cnt);
  finalize_kernel<<<blkGD, 256, 0, stream>>>(sumA, sqA, cnt, n1_ms, meanA, invA);
  norm_apply_kernel<<<blkND, 256, 0, stream>>>(x, batch, meanA, invA, n1_w, n1_b, h1);

  // SAGEConv max-aggregation
  scatter_max_kernel<<<blkEdge, 256, 0, stream>>>(h1, srcI, dstI, agg);

  // GEMMs (WMMA) + bias + residual + relu, fused with norm-2 stats
  gemm_res_relu_stats<<<blkGemm, 256, 0, stream>>>(agg, h1, x, batch, w_l, b_l, w_r,
                                                   h2, sumB, sqB);

  // GraphNorm 2 -> d_out
  finalize_kernel<<<blkGD, 256, 0, stream>>>(sumB, sqB, cnt, n2_ms, meanB, invB);
  norm_apply_kernel<<<blkND, 256, 0, stream>>>(h2, batch, meanB, invB, n2_w, n2_b,
                                               (float*)d_out);
}